// TopologyPreservingLoss_8624294331226
// MI455X (gfx1250) — compile-verified
//
#include <hip/hip_runtime.h>
#include <hip/hip_bf16.h>

#define Bn 2
#define Hn 512
#define Wn 512
#define HWn (Hn*Wn)
#define Ntot (Bn*Hn*Wn)
#define NPART 512
#define INFV 10000.0f

typedef __attribute__((ext_vector_type(2))) float v2f;
typedef __attribute__((ext_vector_type(8))) float v8f;

__device__ __forceinline__ float sigm(float x) {
  if (x >= 0.0f) { return 1.0f / (1.0f + expf(-x)); }
  float e = expf(x); return e / (1.0f + e);
}
__device__ __forceinline__ float bce1(float x, float t) {
  return fmaxf(x, 0.0f) - x * t + log1pf(expf(-fabsf(x)));
}

// ---------------- Stage 1: sigmoid + dice/bce partial sums ----------------
__global__ void k_stage1(const float* __restrict__ pred, const float* __restrict__ targ,
                         float* __restrict__ pbuf, float* __restrict__ part) {
  __shared__ float sm[4][256];
  float sp = 0.f, st = 0.f, spt = 0.f, sb = 0.f;
  for (int i = blockIdx.x * blockDim.x + threadIdx.x; i < Ntot; i += gridDim.x * blockDim.x) {
    float x = pred[i], t = targ[i];
    float p = sigm(x);
    pbuf[i] = p;
    sp += p; st += t; spt += p * t;
    sb += bce1(x, t);
  }
  int tid = threadIdx.x;
  sm[0][tid] = sp; sm[1][tid] = st; sm[2][tid] = spt; sm[3][tid] = sb;
  __syncthreads();
  for (int s = 128; s > 0; s >>= 1) {
    if (tid < s) {
      #pragma unroll
      for (int q = 0; q < 4; ++q) sm[q][tid] += sm[q][tid + s];
    }
    __syncthreads();
  }
  if (tid == 0) {
    #pragma unroll
    for (int q = 0; q < 4; ++q) part[q * NPART + blockIdx.x] = sm[q][0];
  }
}

// ---------------- 3x3 min/max pool (valid-window, matches -inf padded reduce) ----------------
__global__ void k_pool3(const float* __restrict__ in, float* __restrict__ out, int isMax) {
  int i = blockIdx.x * blockDim.x + threadIdx.x;
  if (i >= Ntot) return;
  int x = i & (Wn - 1);
  int y = (i >> 9) & (Hn - 1);
  float r = isMax ? -3.4e38f : 3.4e38f;
  #pragma unroll
  for (int dy = -1; dy <= 1; ++dy) {
    int yy = y + dy; if (yy < 0 || yy >= Hn) continue;
    #pragma unroll
    for (int dx = -1; dx <= 1; ++dx) {
      int xx = x + dx; if (xx < 0 || xx >= Wn) continue;
      float v = in[i + dy * Wn + dx];
      r = isMax ? fmaxf(r, v) : fminf(r, v);
    }
  }
  out[i] = r;
}

__global__ void k_skel_update(float* __restrict__ skel, const float* __restrict__ x,
                              const float* __restrict__ mx, int init) {
  int i = blockIdx.x * blockDim.x + threadIdx.x;
  if (i >= Ntot) return;
  float b = x[i] - mx[i];
  skel[i] = init ? b : fmaxf(skel[i], b);
}

// ---------------- Skeleton correlation partial sums ----------------
__global__ void k_skelsums(const float* __restrict__ skp, const float* __restrict__ skt,
                           const float* __restrict__ targ, const float* __restrict__ pbuf,
                           float* __restrict__ part) {
  __shared__ float sm[4][256];
  float a0 = 0.f, a1 = 0.f, a2 = 0.f, a3 = 0.f;
  for (int i = blockIdx.x * blockDim.x + threadIdx.x; i < Ntot; i += gridDim.x * blockDim.x) {
    float sp_ = skp[i], st_ = skt[i];
    a0 += sp_ * targ[i];
    a1 += sp_;
    a2 += st_ * pbuf[i];
    a3 += st_;
  }
  int tid = threadIdx.x;
  sm[0][tid] = a0; sm[1][tid] = a1; sm[2][tid] = a2; sm[3][tid] = a3;
  __syncthreads();
  for (int s = 128; s > 0; s >>= 1) {
    if (tid < s) {
      #pragma unroll
      for (int q = 0; q < 4; ++q) sm[q][tid] += sm[q][tid + s];
    }
    __syncthreads();
  }
  if (tid == 0) {
    #pragma unroll
    for (int q = 0; q < 4; ++q) part[q * NPART + blockIdx.x] = sm[q][0];
  }
}

// ---------------- EDT vertical two-way scan (per column of 4 masks) ----------------
__global__ void k_edt_vert(const float* __restrict__ targ, float* __restrict__ g2) {
  int idx = blockIdx.x * blockDim.x + threadIdx.x;
  if (idx >= 4 * Wn) return;
  int n = idx / Wn, x = idx - n * Wn;
  int b = n & 1;
  int inv = (n >= 2);
  const float* m = targ + b * HWn;
  float* g = g2 + n * HWn;
  float carry = INFV;
  for (int y = 0; y < Hn; ++y) {
    float tv = m[y * Wn + x];
    bool z = inv ? (tv == 1.0f) : (tv == 0.0f);
    float d = z ? 0.0f : carry + 1.0f;
    carry = d;
    g[y * Wn + x] = d;
  }
  carry = INFV;
  for (int y = Hn - 1; y >= 0; --y) {
    float tv = m[y * Wn + x];
    bool z = inv ? (tv == 1.0f) : (tv == 0.0f);
    float d = z ? 0.0f : carry + 1.0f;
    carry = d;
    float gg = fminf(g[y * Wn + x], d);
    g[y * Wn + x] = gg * gg;
  }
}

// ---------------- EDT horizontal brute-force min-plus + sqrt (per row) ----------------
__global__ void k_edt_horiz(float* __restrict__ g2) {
  __shared__ float row[Wn];
  float* g = g2 + (size_t)blockIdx.x * Wn;
  for (int j = threadIdx.x; j < Wn; j += blockDim.x) row[j] = g[j];
  __syncthreads();
  for (int x = threadIdx.x; x < Wn; x += blockDim.x) {
    float best = 3.4e38f;
    for (int j = 0; j < Wn; ++j) {
      float diff = (float)(x - j);
      best = fminf(best, row[j] + diff * diff);
    }
    g[x] = sqrtf(best);
  }
}

// ---------------- global max of wmap = exp(-min(dfg,dbg)/theta0) ----------------
__global__ void k_wmax(const float* __restrict__ dist, float* __restrict__ part) {
  __shared__ float sm[256];
  float mx = 0.0f;
  for (int i = blockIdx.x * blockDim.x + threadIdx.x; i < Ntot; i += gridDim.x * blockDim.x) {
    float dm = fminf(dist[i], dist[i + 2 * HWn]);
    mx = fmaxf(mx, expf(-dm / 3.0f));
  }
  int tid = threadIdx.x;
  sm[tid] = mx;
  __syncthreads();
  for (int s = 128; s > 0; s >>= 1) {
    if (tid < s) sm[tid] = fmaxf(sm[tid], sm[tid + s]);
    __syncthreads();
  }
  if (tid == 0) part[blockIdx.x] = sm[0];
}

__global__ void k_maxfin(const float* __restrict__ part, float* __restrict__ dst) {
  float m = part[0];
  for (int i = 1; i < NPART; ++i) m = fmaxf(m, part[i]);
  dst[0] = m;
}

// ---------------- boundary + small-vessel weighted BCE partial sums ----------------
__global__ void k_finalsums(const float* __restrict__ pred, const float* __restrict__ targ,
                            const float* __restrict__ dist, const float* __restrict__ scal,
                            float* __restrict__ part) {
  __shared__ float sm[2][256];
  float maxw = scal[8];
  float a0 = 0.f, a1 = 0.f;
  for (int i = blockIdx.x * blockDim.x + threadIdx.x; i < Ntot; i += gridDim.x * blockDim.x) {
    float x = pred[i], t = targ[i];
    float b = bce1(x, t);
    float dfg = dist[i];
    float dm = fminf(dfg, dist[i + 2 * HWn]);
    float fw = 1.0f + 5.0f * (expf(-dm / 3.0f) / maxw);
    float width = 2.0f * dfg;
    float w2 = (width > 0.0f && width < 5.0f) ? 3.0f : 1.0f;
    a0 += b * fw;
    a1 += b * w2;
  }
  int tid = threadIdx.x;
  sm[0][tid] = a0; sm[1][tid] = a1;
  __syncthreads();
  for (int s = 128; s > 0; s >>= 1) {
    if (tid < s) { sm[0][tid] += sm[0][tid + s]; sm[1][tid] += sm[1][tid + s]; }
    __syncthreads();
  }
  if (tid == 0) {
    part[0 * NPART + blockIdx.x] = sm[0][0];
    part[1 * NPART + blockIdx.x] = sm[1][0];
  }
}

// ---------------- WMMA f32 16x16x4 finalize: sum 512 partials per quantity ----------------
// One wave (32 threads), EXEC all ones. A = 64-partial chunk (16x4 f32), B = ones(4x16),
// C accumulates; D[m][n] = sum_k A[m][k] + C[m][n]. Total = sum_m D[m][*].
__global__ void k_finalize_wmma(const float* __restrict__ part, float* __restrict__ dst, int nq) {
  int lane = threadIdx.x;
  int m = lane & 15;
  int k0 = (lane >> 4) << 1;   // 0 for lanes 0-15, 2 for lanes 16-31
  v2f bm; bm.x = 1.0f; bm.y = 1.0f;
  for (int q = 0; q < nq; ++q) {
    const float* base = part + q * NPART;
    v8f c = {};
    for (int ch = 0; ch < NPART / 64; ++ch) {
      const float* pc = base + ch * 64;
      v2f a;
      a.x = pc[m * 4 + k0];
      a.y = pc[m * 4 + k0 + 1];
      c = __builtin_amdgcn_wmma_f32_16x16x4_f32(false, a, false, bm, (short)0, c,
                                                false, false);
    }
    float s = c[0] + c[1] + c[2] + c[3] + c[4] + c[5] + c[6] + c[7];
    s += __shfl_xor(s, 16, 32);   // combine M=0..7 half with M=8..15 half
    if (lane == 0) dst[q] = s;
  }
}

// ---------------- combine scalar terms ----------------
__global__ void k_combine(const float* __restrict__ s, float* __restrict__ out) {
  float sp = s[0], st = s[1], spt = s[2], sb = s[3];
  float spk_t = s[4], spk = s[5], stk_p = s[6], stk = s[7];
  float sbfw = s[9], sbw2 = s[10];
  float invN = 1.0f / (float)Ntot;
  float dice = 1.0f - (2.0f * spt + 1.0f) / (sp + st + 1.0f);
  float bce = sb * invN;
  float seg = (0.5f / 0.7f) * dice + (0.2f / 0.7f) * bce;
  float tprec = spk_t / (spk + 1.0f);
  float tsens = stk_p / (stk + 1.0f);
  float topo = 1.0f - 2.0f * tprec * tsens / (tprec + tsens + 1.0f);
  float boundary = sbfw * invN;
  float sv = sbw2 * invN;
  out[0] = seg + 0.1f * topo + 0.15f * boundary + 0.1f * sv;   // W_HD * 0 omitted
}

extern "C" void kernel_launch(void* const* d_in, const int* in_sizes, int n_in,
                              void* d_out, int out_size, void* d_ws, size_t ws_size,
                              hipStream_t stream) {
  const float* pred = (const float*)d_in[0];
  const float* targ = (const float*)d_in[1];
  float* out = (float*)d_out;
  float* Wks = (float*)d_ws;

  float* scal  = Wks;                    // 16 scalar slots
  float* partA = Wks + 64;               // 4*NPART
  float* partB = partA + 4 * NPART;      // 4*NPART
  float* partC = partB + 4 * NPART;      // NPART
  float* partD = partC + NPART;          // 2*NPART
  float* pbuf  = Wks + 8192;             // Ntot
  float* skelP = pbuf + Ntot;            // Ntot
  float* skelT = skelP + Ntot;           // Ntot
  float* S1    = skelT + Ntot;           // Ntot
  float* S2    = S1 + Ntot;              // Ntot
  float* MXb   = S2 + Ntot;              // Ntot
  float* edt   = S1;                     // reuse S1+S2 (2*Ntot = 4*H*W) after skeleton

  const int TPB = 256;
  const int EW_BLOCKS = (Ntot + TPB - 1) / TPB;   // 2048

  // Stage 1: sigmoid + dice/bce sums
  k_stage1<<<NPART, TPB, 0, stream>>>(pred, targ, pbuf, partA);
  k_finalize_wmma<<<1, 32, 0, stream>>>(partA, scal + 0, 4);

  // Soft skeletonization: p channel then t channel (shared scratch)
  for (int ch = 0; ch < 2; ++ch) {
    const float* x = (ch == 0) ? pbuf : targ;
    float* skel = (ch == 0) ? skelP : skelT;
    float* a = S1; float* b = S2;
    for (int it = 0; it < 5; ++it) {
      k_pool3<<<EW_BLOCKS, TPB, 0, stream>>>(x, a, 0);      // erode  -> a
      k_pool3<<<EW_BLOCKS, TPB, 0, stream>>>(a, MXb, 1);    // dilate -> MXb
      k_skel_update<<<EW_BLOCKS, TPB, 0, stream>>>(skel, x, MXb, it == 0);
      x = a; float* tp = a; a = b; b = tp;
    }
  }
  k_skelsums<<<NPART, TPB, 0, stream>>>(skelP, skelT, targ, pbuf, partB);
  k_finalize_wmma<<<1, 32, 0, stream>>>(partB, scal + 4, 4);

  // EDT on {t0, t1, 1-t0, 1-t1}
  k_edt_vert<<<(4 * Wn + TPB - 1) / TPB, TPB, 0, stream>>>(targ, edt);
  k_edt_horiz<<<4 * Hn, TPB, 0, stream>>>(edt);

  // Global max of wmap, then weighted BCE sums
  k_wmax<<<NPART, TPB, 0, stream>>>(edt, partC);
  k_maxfin<<<1, 1, 0, stream>>>(partC, scal + 8);
  k_finalsums<<<NPART, TPB, 0, stream>>>(pred, targ, edt, scal, partD);
  k_finalize_wmma<<<1, 32, 0, stream>>>(partD, scal + 9, 2);

  k_combine<<<1, 1, 0, stream>>>(scal, out);
}